// DKD_61495341744275
// MI455X (gfx1250) — compile-verified
//
#include <hip/hip_runtime.h>

// ---------------- problem constants ----------------
#define Hh 1536
#define Ww 1536
#define Cc 64
#define RAD 2
#define KER 4
#define THn (Hh / KER)          // 384
#define TWn (Ww / KER)          // 384
#define NT (THn * TWn)          // 147456
#define TOPK 500
#define NBIN 65536
// output layout: kpts [500*2] | descriptors [500*64] | scores [500]
#define OUT_DESC 1000
#define OUT_SC   33000

typedef __attribute__((ext_vector_type(2))) float v2f;
typedef __attribute__((ext_vector_type(8))) float v8f;

// ---------------- kernel 1: zero hist/ctrs/list ----------------
__global__ void dkd_zero_kernel(unsigned* __restrict__ p, int n) {
    int i = blockIdx.x * blockDim.x + threadIdx.x;
    if (i < n) p[i] = 0u;
}

// ---------------- kernel 2: 4x4 tile max + hist1 ----------------
__global__ __launch_bounds__(256) void dkd_tilemax_kernel(
    const float* __restrict__ s, unsigned* __restrict__ keys,
    unsigned* __restrict__ pos, unsigned* __restrict__ hist1) {
    int t = blockIdx.x * blockDim.x + threadIdx.x;
    if (t >= NT) return;
    int ty = t / TWn, tx = t % TWn;
    int r0 = ty * KER, c0 = tx * KER;
    float best = -1.0f;
    int barg = 0;
    for (int kr = 0; kr < KER; ++kr) {
        int r = r0 + kr;
        float4 v = *(const float4*)(s + (size_t)r * Ww + c0);
        float e0 = v.x, e1 = v.y, e2 = v.z, e3 = v.w;
        bool rowB = (r <= RAD) || (r >= Hh - RAD);
        float e[4] = {e0, e1, e2, e3};
        for (int kc = 0; kc < KER; ++kc) {
            int cc = c0 + kc;
            float val = (rowB || (cc <= RAD) || (cc >= Ww - RAD)) ? 0.0f : e[kc];
            if (val > best) { best = val; barg = kr * KER + kc; }  // first max kept
        }
    }
    unsigned key = __float_as_uint(best);            // scores >= 0 -> monotone
    int gy = r0 + (barg >> 2), gx = c0 + (barg & 3);
    keys[t] = key;
    pos[t] = ((unsigned)gy << 16) | (unsigned)gx;
    atomicAdd(&hist1[key >> 16], 1u);
}

// ---------------- kernels 3/5: radix-select bin finder ----------------
// mode 0: need=TOPK on hist1 -> ctrs[0]=b1, ctrs[1]=strictlyAbove
// mode 1: need=TOPK-ctrs[1] on hist2 -> ctrs[2]=T, ctrs[3]=needEq
__global__ __launch_bounds__(1024) void dkd_find_kernel(
    const unsigned* __restrict__ hist, unsigned* __restrict__ ctrs, int mode) {
    __shared__ unsigned S[1024];
    int tid = threadIdx.x;
    unsigned acc = 0;
    const unsigned* hb = hist + tid * 64;
    for (int j = 0; j < 64; ++j) acc += hb[j];
    S[tid] = acc;
    __syncthreads();
    if (tid == 0) {
        unsigned need = (mode == 0) ? (unsigned)TOPK : ((unsigned)TOPK - ctrs[1]);
        unsigned cum = 0;
        int seg = 0;
        for (int i = 1023; i >= 0; --i) {
            if (cum + S[i] >= need) { seg = i; break; }
            cum += S[i];
        }
        int bstar = seg * 64;
        for (int b = seg * 64 + 63; b >= seg * 64; --b) {
            unsigned h = hist[b];
            if (cum + h >= need) { bstar = b; break; }
            cum += h;
        }
        if (mode == 0) {
            ctrs[0] = (unsigned)bstar;  // b1
            ctrs[1] = cum;              // count strictly above bin b1
        } else {
            ctrs[2] = (ctrs[0] << 16) | (unsigned)bstar;  // exact threshold key T
            ctrs[3] = need - cum;                          // #keys == T to take
        }
    }
}

// ---------------- kernel 4: second-level histogram ----------------
__global__ __launch_bounds__(256) void dkd_hist2_kernel(
    const unsigned* __restrict__ keys, const unsigned* __restrict__ ctrs,
    unsigned* __restrict__ hist2) {
    unsigned b1 = ctrs[0];
    int t = blockIdx.x * blockDim.x + threadIdx.x;
    if (t < NT) {
        unsigned k = keys[t];
        if ((k >> 16) == b1) atomicAdd(&hist2[k & 0xFFFFu], 1u);
    }
}

// ---------------- kernel 6: compact exact top-500 ----------------
__global__ __launch_bounds__(256) void dkd_compact_kernel(
    const unsigned* __restrict__ keys, unsigned* __restrict__ ctrs,
    uint2* __restrict__ list) {
    unsigned T = ctrs[2], needEq = ctrs[3];
    int t = blockIdx.x * blockDim.x + threadIdx.x;
    if (t < NT) {
        unsigned k = keys[t];
        if (k > T) {
            unsigned slot = atomicAdd(&ctrs[4], 1u);
            list[slot] = make_uint2(k, (unsigned)t);
        } else if (k == T) {
            unsigned e = atomicAdd(&ctrs[5], 1u);
            if (e < needEq) {
                unsigned slot = atomicAdd(&ctrs[4], 1u);
                list[slot] = make_uint2(k, (unsigned)t);
            }
        }
    }
}

// ---------------- kernel 7: bitonic sort 512 (ascending, stable) ----------------
__global__ __launch_bounds__(512) void dkd_sort_kernel(uint2* __restrict__ list) {
    __shared__ unsigned sk[512];
    __shared__ unsigned si[512];
    int tid = threadIdx.x;
    uint2 e = list[tid];               // entries 500..511 are (0,0) padding
    sk[tid] = e.x; si[tid] = e.y;
    for (int k = 2; k <= 512; k <<= 1) {
        for (int j = k >> 1; j > 0; j >>= 1) {
            __syncthreads();
            int ixj = tid ^ j;
            if (ixj > tid) {
                unsigned ka = sk[tid], kb = sk[ixj];
                unsigned ia = si[tid], ib = si[ixj];
                bool agtb = (ka > kb) || (ka == kb && ia > ib);
                bool up = ((tid & k) == 0);
                if (agtb == up) {
                    sk[tid] = kb; sk[ixj] = ka;
                    si[tid] = ib; si[ixj] = ia;
                }
            }
        }
    }
    __syncthreads();
    list[tid] = make_uint2(sk[tid], si[tid]);
}

// ---------------- kernel 8: keypoints + scores ----------------
__global__ void dkd_kpts_kernel(const uint2* __restrict__ list,
                                const unsigned* __restrict__ pos,
                                float* __restrict__ out) {
    int k = blockIdx.x * blockDim.x + threadIdx.x;
    if (k < TOPK) {
        uint2 e = list[12 + k];        // entries 12..511 = ascending top-500
        unsigned p = pos[e.y];
        out[2 * k + 0] = (float)(p & 0xFFFFu);   // x = col
        out[2 * k + 1] = (float)(p >> 16);       // y = row
        out[OUT_SC + k] = __uint_as_float(e.x);  // score
    }
}

// ---------------- kernel 9: descriptor gather + WMMA Gram-diagonal norm ----
// One wave (32 lanes) per 16 keypoints. Norms^2 = diag(A * A^T) computed by
// V_WMMA_F32_16X16X4_F32: A rows and B columns share the same per-lane VGPR
// layout, so passing the same v2f pair as A and B yields the Gram matrix.
__global__ __launch_bounds__(32) void dkd_desc_kernel(
    const float* __restrict__ desc, const uint2* __restrict__ list,
    const unsigned* __restrict__ pos, float* __restrict__ out) {
    __shared__ float raw[16][64];
    __shared__ float invn[16];
    const int lane = threadIdx.x;          // 0..31
    const int base = blockIdx.x * 16;

    for (int i = 0; i < 16; ++i) {
        int k = base + i;
        int kk = (k < TOPK) ? k : (TOPK - 1);
        uint2 e = list[12 + kk];
        unsigned p = pos[e.y];
        size_t off = (size_t)(p >> 16) * Ww + (size_t)(p & 0xFFFFu);
        raw[i][lane]      = desc[(size_t)lane * (Hh * Ww) + off];
        raw[i][lane + 32] = desc[(size_t)(lane + 32) * (Hh * Ww) + off];
    }
    __syncthreads();

    const int m = lane & 15;               // matrix row held by this lane
    const int kh = lane >> 4;              // K half (lanes 16-31 hold K=2,3)
    v8f c = {};
    for (int s = 0; s < 16; ++s) {         // 16 steps x K=4 = 64 channels
        v2f a;
        a.x = raw[m][4 * s + 2 * kh + 0];
        a.y = raw[m][4 * s + 2 * kh + 1];
        c = __builtin_amdgcn_wmma_f32_16x16x4_f32(
                /*neg_a=*/false, a, /*neg_b=*/false, a,
                /*c_mod=*/(short)0, c, /*reuse_a=*/false, /*reuse_b=*/false);
    }
    // diag(m): m<8 -> (VGPR m, lane m); m>=8 -> (VGPR m-8, lane m+16)
    if (lane < 8) {
        invn[lane] = 1.0f / sqrtf(c[lane]);
    } else if (lane >= 24) {
        invn[lane - 16] = 1.0f / sqrtf(c[lane - 24]);
    }
    __syncthreads();

    for (int i = 0; i < 16; ++i) {
        int k = base + i;
        if (k < TOPK) {
            float sc = invn[i];
            out[OUT_DESC + (size_t)k * Cc + lane]      = raw[i][lane] * sc;
            out[OUT_DESC + (size_t)k * Cc + lane + 32] = raw[i][lane + 32] * sc;
        }
    }
}

// ---------------- launcher ----------------
extern "C" void kernel_launch(void* const* d_in, const int* in_sizes, int n_in,
                              void* d_out, int out_size, void* d_ws, size_t ws_size,
                              hipStream_t stream) {
    (void)in_sizes; (void)n_in; (void)out_size; (void)ws_size;
    const float* scores = (const float*)d_in[0];   // [1,1,1536,1536]
    const float* desc   = (const float*)d_in[1];   // [1,64,1536,1536]
    float* out = (float*)d_out;                    // 33500 floats

    // workspace layout (uints)
    unsigned* ws    = (unsigned*)d_ws;
    unsigned* keys  = ws;                  // NT
    unsigned* pos   = ws + NT;             // NT
    unsigned* hist1 = ws + 2 * NT;         // 65536
    unsigned* hist2 = hist1 + NBIN;        // 65536
    unsigned* ctrs  = hist2 + NBIN;        // 16
    uint2*    list  = (uint2*)(ctrs + 16); // 512 pairs

    const int nzero = 2 * NBIN + 16 + 1024;  // hist1+hist2+ctrs+list
    dkd_zero_kernel<<<(nzero + 255) / 256, 256, 0, stream>>>(hist1, nzero);

    dkd_tilemax_kernel<<<NT / 256, 256, 0, stream>>>(scores, keys, pos, hist1);
    dkd_find_kernel<<<1, 1024, 0, stream>>>(hist1, ctrs, 0);
    dkd_hist2_kernel<<<NT / 256, 256, 0, stream>>>(keys, ctrs, hist2);
    dkd_find_kernel<<<1, 1024, 0, stream>>>(hist2, ctrs, 1);
    dkd_compact_kernel<<<NT / 256, 256, 0, stream>>>(keys, ctrs, list);
    dkd_sort_kernel<<<1, 512, 0, stream>>>(list);
    dkd_kpts_kernel<<<2, 256, 0, stream>>>(list, pos, out);
    dkd_desc_kernel<<<(TOPK + 15) / 16, 32, 0, stream>>>(desc, list, pos, out);
}